// GCN_37426345017679
// MI455X (gfx1250) — compile-verified
//
#include <hip/hip_runtime.h>
#include <hip/hip_bf16.h>

// ---------------------------------------------------------------------------
// GCN 2-layer forward for gfx1250 (MI455X).
//   L1: h = relu( Dinv(A+I)Dinv (x@W1) + b1 )   128 -> 16   (WMMA f32 16x16x4)
//   L2: out =     Dinv(A+I)Dinv (h@W2) + b2     16  -> 2    (scalar FMA)
// ---------------------------------------------------------------------------

typedef __attribute__((ext_vector_type(2))) float v2f;
typedef __attribute__((ext_vector_type(8))) float v8f;

#define HIDC 16
#define INC  128
#define CDIV(a, b) (((a) + (b) - 1) / (b))

// ---------------- degree / normalization -----------------------------------

__global__ void k_set_one(float* __restrict__ deg, int n) {
    int i = blockIdx.x * blockDim.x + threadIdx.x;
    if (i < n) deg[i] = 1.0f;   // self-loop contributes 1
}

__global__ void k_deg_edges(const long long* __restrict__ dst,
                            float* __restrict__ deg, int e) {
    int i = blockIdx.x * blockDim.x + threadIdx.x;
    if (i < e) atomicAdd(&deg[(int)dst[i]], 1.0f);
}

__global__ void k_rsqrt_inplace(float* __restrict__ deg, int n) {
    int i = blockIdx.x * blockDim.x + threadIdx.x;
    if (i < n) deg[i] = rsqrtf(deg[i]);   // deg >= 1 always (self-loop)
}

// ---------------- layer-1 GEMM: h = x @ W1 via V_WMMA_F32_16X16X4_F32 ------
// One wave per 16x16 output tile; 8 waves (256 threads) per block -> 128 rows.
// A (16x4 f32): lane L: m = L%16, kpair = (L/16)*2 -> {A[m][k+kp], A[m][k+kp+1]}
// B (4x16 f32): lane L: n = L%16, kpair = (L/16)*2 -> {B[k+kp][n], B[k+kp+1][n]}
// C/D (16x16) : vgpr r: m = r + 8*(L/16), n = L%16

__global__ __launch_bounds__(256) void k_gemm1_wmma(
        const float* __restrict__ x, const float* __restrict__ W1,
        float* __restrict__ h, int n) {
    __shared__ float sW[INC * HIDC];           // 8 KB of the 320 KB WGP LDS
    for (int i = threadIdx.x; i < INC * HIDC; i += 256) sW[i] = W1[i];
    __syncthreads();

    const int wave = threadIdx.x >> 5;
    const int lane = threadIdx.x & 31;
    const int row0 = (blockIdx.x * 8 + wave) * 16;     // wave-uniform
    if (row0 < n) {                                    // uniform -> EXEC all 1s
        const int m  = lane & 15;
        const int kp = (lane >> 4) << 1;               // 0 or 2
        const int nn = lane & 15;

        const float* __restrict__ xr = x + (size_t)(row0 + m) * INC;

        v8f c = {};
        #pragma unroll
        for (int k = 0; k < INC; k += 4) {
            v2f a, b;
            a.x = xr[k + kp];
            a.y = xr[k + kp + 1];
            b.x = sW[(k + kp) * HIDC + nn];
            b.y = sW[(k + kp + 1) * HIDC + nn];
            c = __builtin_amdgcn_wmma_f32_16x16x4_f32(
                    false, a, false, b, (short)0, c, false, false);
        }

        const int mo = (lane >> 4) << 3;               // 0 or 8
        #pragma unroll
        for (int r = 0; r < 8; ++r) {
            int rr = row0 + mo + r;
            if (rr < n) h[(size_t)rr * HIDC + nn] = c[r];
        }
    }
}

// ---------------- layer-1 aggregation --------------------------------------

// acc[i][c] = h[i][c] * dinv[i]^2   (self-loop term; also fully inits buffer)
__global__ void k_agg1_init(const float* __restrict__ h,
                            const float* __restrict__ dinv,
                            float* __restrict__ acc, int n) {
    size_t t = (size_t)blockIdx.x * blockDim.x + threadIdx.x;
    if (t < (size_t)n * HIDC) {
        int i = (int)(t >> 4);
        float w = dinv[i] * dinv[i];
        acc[t] = h[t] * w;
    }
}

// one thread per (edge, channel): 16 lanes share an edge -> coalesced 64B gather
__global__ void k_agg1_edges(const long long* __restrict__ src,
                             const long long* __restrict__ dst,
                             const float* __restrict__ h,
                             const float* __restrict__ dinv,
                             float* __restrict__ acc, long long e16) {
    size_t t = (size_t)blockIdx.x * blockDim.x + threadIdx.x;
    if (t >= (size_t)e16) return;
    size_t e = t >> 4;
    int    c = (int)(t & 15);
    int s = (int)src[e];
    int d = (int)dst[e];
    float w = dinv[s] * dinv[d];
    atomicAdd(&acc[(size_t)d * HIDC + c], h[(size_t)s * HIDC + c] * w);
}

// ---------------- fused bias + ReLU + layer-2 GEMM (16 -> 2) ---------------

__global__ void k_bias_relu_gemm2(const float* __restrict__ acc,
                                  const float* __restrict__ b1,
                                  const float* __restrict__ W2,
                                  float* __restrict__ h2, int n) {
    int i = blockIdx.x * blockDim.x + threadIdx.x;
    if (i >= n) return;
    float c0 = 0.f, c1 = 0.f;
    #pragma unroll
    for (int k = 0; k < HIDC; ++k) {
        float v = acc[(size_t)i * HIDC + k] + b1[k];
        v = fmaxf(v, 0.f);
        c0 = fmaf(v, W2[k * 2 + 0], c0);
        c1 = fmaf(v, W2[k * 2 + 1], c1);
    }
    h2[(size_t)i * 2 + 0] = c0;
    h2[(size_t)i * 2 + 1] = c1;
}

// ---------------- layer-2 aggregation --------------------------------------

// out[i][c] = b2[c] + h2[i][c] * dinv[i]^2   (bias + self-loop; inits d_out)
__global__ void k_agg2_init(const float* __restrict__ h2,
                            const float* __restrict__ dinv,
                            const float* __restrict__ b2,
                            float* __restrict__ out, int n) {
    int i = blockIdx.x * blockDim.x + threadIdx.x;
    if (i >= n) return;
    float w = dinv[i] * dinv[i];
    out[(size_t)i * 2 + 0] = b2[0] + h2[(size_t)i * 2 + 0] * w;
    out[(size_t)i * 2 + 1] = b2[1] + h2[(size_t)i * 2 + 1] * w;
}

__global__ void k_agg2_edges(const long long* __restrict__ src,
                             const long long* __restrict__ dst,
                             const float* __restrict__ h2,
                             const float* __restrict__ dinv,
                             float* __restrict__ out, int e) {
    int i = blockIdx.x * blockDim.x + threadIdx.x;
    if (i >= e) return;
    int s = (int)src[i];
    int d = (int)dst[i];
    float w = dinv[s] * dinv[d];
    atomicAdd(&out[(size_t)d * 2 + 0], h2[(size_t)s * 2 + 0] * w);
    atomicAdd(&out[(size_t)d * 2 + 1], h2[(size_t)s * 2 + 1] * w);
}

// ---------------------------------------------------------------------------

extern "C" void kernel_launch(void* const* d_in, const int* in_sizes, int n_in,
                              void* d_out, int out_size, void* d_ws, size_t ws_size,
                              hipStream_t stream) {
    (void)n_in; (void)out_size; (void)ws_size;

    const float*     x   = (const float*)d_in[0];
    const long long* ei  = (const long long*)d_in[1];
    const float*     W1  = (const float*)d_in[2];
    const float*     b1  = (const float*)d_in[3];
    const float*     W2  = (const float*)d_in[4];
    const float*     b2  = (const float*)d_in[5];
    float*           out = (float*)d_out;

    const int n = in_sizes[0] / INC;        // 200000
    const int e = in_sizes[1] / 2;          // 6400000
    const long long* src = ei;
    const long long* dst = ei + e;

    // workspace layout (floats)
    float* ws   = (float*)d_ws;
    float* h    = ws;                        // n*16
    float* acc  = h   + (size_t)n * HIDC;    // n*16
    float* h2   = acc + (size_t)n * HIDC;    // n*2
    float* dinv = h2  + (size_t)n * 2;       // n   (deg, then rsqrt in place)

    const int B = 256;

    // normalization: deg -> dinv
    k_set_one     <<<CDIV(n, B), B, 0, stream>>>(dinv, n);
    k_deg_edges   <<<CDIV(e, B), B, 0, stream>>>(dst, dinv, e);
    k_rsqrt_inplace<<<CDIV(n, B), B, 0, stream>>>(dinv, n);

    // layer 1: WMMA GEMM, then normalized aggregation
    k_gemm1_wmma  <<<CDIV(CDIV(n, 16), 8), B, 0, stream>>>(x, W1, h, n);
    k_agg1_init   <<<CDIV((long long)n * HIDC, B), B, 0, stream>>>(h, dinv, acc, n);
    k_agg1_edges  <<<CDIV((long long)e * HIDC, B), B, 0, stream>>>(
                      src, dst, h, dinv, acc, (long long)e * HIDC);

    // bias + relu + layer-2 transform
    k_bias_relu_gemm2<<<CDIV(n, B), B, 0, stream>>>(acc, b1, W2, h2, n);

    // layer 2 aggregation into d_out
    k_agg2_init   <<<CDIV(n, B), B, 0, stream>>>(h2, dinv, b2, out, n);
    k_agg2_edges  <<<CDIV(e, B), B, 0, stream>>>(src, dst, h2, dinv, out, e);
}